// VectorizedTrajectoryFreeLoss_19181323944318
// MI455X (gfx1250) — compile-verified
//
#include <hip/hip_runtime.h>
#include <cmath>

// ---------------------------------------------------------------------------
// VectorizedTrajectoryFreeLoss for MI455X (gfx1250, wave32)
//   - pairwise MLP (dominant cost): lane-parallel v_tanh_f32, j-partial-sums
//     accumulated on the matrix pipe via v_wmma_f32_16x16x4_f32 with an
//     all-ones B (D[i][*] += sum_k A[i][k]).
//   - particle tile (2560 B) staged to LDS with global_load_async_to_lds_b128
//     + s_wait_asynccnt; inner loop is pure LDS + VALU + WMMA.
//   - j-loop split 4 ways per (b,i-tile); partials combined with
//     global_atomic_add_f32 into zero-inited accumulators.
// ---------------------------------------------------------------------------

typedef float v2f __attribute__((ext_vector_type(2)));
typedef float v8f __attribute__((ext_vector_type(8)));

#if defined(__has_builtin)
#  if __has_builtin(__builtin_amdgcn_tanhf)
#    define TANHF(x) __builtin_amdgcn_tanhf(x)
#    pragma message("CDNA5 probe: tanh -> __builtin_amdgcn_tanhf (v_tanh_f32)")
#  endif
#endif
#ifndef TANHF
#  define TANHF(x) tanhf(x)
#  pragma message("CDNA5 probe: tanh -> libm tanhf fallback")
#endif

#if defined(__has_builtin)
#  if __has_builtin(__builtin_amdgcn_rsqf)
#    define RSQF(x) __builtin_amdgcn_rsqf(x)
#  endif
#endif
#ifndef RSQF
#  define RSQF(x) (1.0f / sqrtf(x))
#endif

#define Mm 4
#define Ll 6
#define Nn 320
#define Dd 2
#define Hh 32
#define Bb (Mm * (Ll - 1))   // 20
#define NT (Nn / 16)         // 20 i-tiles
#define JS 4                 // j-segments (Nn/JS = 80 j's each)
#define SIGMA2 0.01f

// float offsets into workspace (each block is Bb*Nn = 6400 floats)
#define BN (Bb * Nn)
#define OFF_PS1  (0 * BN)   // sum_j dPhi*dx/r      (X_curr)  [atomic]
#define OFF_PS2  (1 * BN)   // sum_j dPhi*dy/r      (X_curr)  [atomic]
#define OFF_PS3  (2 * BN)   // sum_j d2Phi+dPhi/r   (X_curr)  [atomic]
#define OFF_PS4  (3 * BN)   // sum_j (Phi+b2)       (X_curr)  [atomic]
#define OFF_PHIN (4 * BN)   // sum_j (Phi+b2)       (X_next)  [atomic]
#define OFF_VC   (5 * BN)   // V(x)   curr
#define OFF_GX   (6 * BN)   // dV/dx0 curr
#define OFF_GY   (7 * BN)   // dV/dx1 curr
#define OFF_LAPV (8 * BN)   // tr Hess V curr
#define OFF_VN   (9 * BN)   // V(x)   next
#define WS_FLOATS (10 * BN) // 64000 floats = 256 KB of d_ws
#define ZERO_FLOATS (5 * BN)

static __device__ __forceinline__ void atomic_addf(float* p, float v) {
  __hip_atomic_fetch_add(p, v, __ATOMIC_RELAXED, __HIP_MEMORY_SCOPE_AGENT);
}

static __device__ __forceinline__ v8f wmma_rowsum(v2f a, v2f ones, v8f c) {
  // D = A(16x4) * ones(4x16) + C  -> every column of row i gets Sigma_k A[i][k]
  return __builtin_amdgcn_wmma_f32_16x16x4_f32(false, a, false, ones,
                                               (short)0, c, false, false);
}

// Async global->LDS copy of one 16B chunk per lane (ISA 10.x / 15.18.3).
// VDST carries the LDS byte address; ASYNCcnt tracks completion.
static __device__ __forceinline__ void async_b128_to_lds(unsigned int lds_addr,
                                                         unsigned long long ga) {
  asm volatile("global_load_async_to_lds_b128 %0, %1, off"
               :: "v"(lds_addr), "v"(ga) : "memory");
}
static __device__ __forceinline__ void wait_asynccnt0() {
  asm volatile("s_wait_asynccnt 0x0" ::: "memory");
}

__global__ void zero_kernel(float* __restrict__ ws) {
  const int tid = blockIdx.x * blockDim.x + threadIdx.x;
  if (tid < ZERO_FLOATS) ws[tid] = 0.0f;
}

// One wave (32 threads) per (b, 16-row i-tile, j-segment). MODE=0: X_curr,
// full derivative pass; MODE=1: X_next, Phi-only energy pass.
template <int MODE>
__global__ __launch_bounds__(32) void pair_kernel(
    const float* __restrict__ data,
    const float* __restrict__ pw1, const float* __restrict__ pb1,
    const float* __restrict__ pw2, const float* __restrict__ pb2,
    float* __restrict__ ws) {
  const int lane = threadIdx.x;            // 0..31, full wave, EXEC all ones
  const int tile = blockIdx.x;
  const int jseg = blockIdx.y;
  const int b = tile / NT;
  const int it = tile % NT;
  const int m = b / (Ll - 1);
  const int lidx = b % (Ll - 1);
  const float* X = data + (size_t)(m * Ll + lidx + MODE) * Nn * Dd;

  // Stage the entire particle tile (320 x float2 = 2560 B) into LDS with
  // async b128 loads: 160 chunks = 5 per lane.
  __shared__ float2 Xs[Nn];
  {
    const unsigned long long gbase = (unsigned long long)(const void*)X;
    // flat->LDS aperture: low 32 bits of a shared pointer are the LDS address
    const unsigned int lbase = (unsigned int)(size_t)(void*)&Xs[0];
#pragma unroll
    for (int k = 0; k < (Nn * Dd * 4) / (16 * 32); ++k) {
      const unsigned int c = lane + 32 * k;
      async_b128_to_lds(lbase + c * 16, gbase + c * 16);
    }
  }

  // Weights in LDS: (w1, b1, w2, w2*w1) packed, plus -2*w2*w1^2.
  __shared__ float4 wp[Hh];
  __shared__ float wc[Hh];
  {
    float w1 = pw1[lane], b1 = pb1[lane], w2 = pw2[lane];
    wp[lane] = make_float4(w1, b1, w2, w2 * w1);
    wc[lane] = -2.0f * w2 * w1 * w1;
  }
  __syncthreads();
  wait_asynccnt0();                        // Xs now valid
  const float phib2 = pb2[0];

  // WMMA 16x16x4 fp32 A-layout: lane l holds row M = l%16, K = vgpr + 2*(l/16)
  const int irow = lane & 15;
  const int hv = lane >> 4;
  const int i = it * 16 + irow;
  const float2 xi = Xs[i];

  v8f c1 = {}, c2 = {}, c3 = {}, c4 = {};
  v2f ones; ones.x = 1.0f; ones.y = 1.0f;   // all-ones B is layout-agnostic

  const int jlo = jseg * (Nn / JS);
  const int jhi = jlo + (Nn / JS);
  for (int jb = jlo; jb < jhi; jb += 4) {
    const int j0 = jb + 2 * hv;             // this lane's two K slots
    float q1[2], q2[2], q3[2], q4[2];
#pragma unroll
    for (int u = 0; u < 2; ++u) {
      const int j = j0 + u;
      const float2 xj = Xs[j];
      const float dx = xi.x - xj.x;
      const float dy = xi.y - xj.y;
      const float sq = dx * dx + dy * dy;
      // r = sq * rsq(sq), inv = rsq(sq); diagonal (sq==0) is masked anyway.
      const float rsq = (sq > 0.0f) ? RSQF(sq) : 0.0f;
      const float r = sq * rsq;
      float aP = 0.0f, aD = 0.0f, aS = 0.0f;
#pragma unroll 8
      for (int h = 0; h < Hh; ++h) {
        const float4 w = wp[h];                 // (w1, b1, w2, w2*w1)
        const float uu = fmaf(r, w.x, w.y);
        const float t = TANHF(uu);
        const float s = fmaf(-t, t, 1.0f);      // 1 - tanh^2
        aP = fmaf(w.z, t, aP);                  // Phi
        if (MODE == 0) {
          aD = fmaf(w.w, s, aD);                // Phi'
          aS = fmaf(wc[h], t * s, aS);          // Phi''
        }
      }
      const float mask = (j == i) ? 0.0f : 1.0f;        // off-diagonal only
      if (MODE == 0) {
        q1[u] = aD * dx * rsq * mask;
        q2[u] = aD * dy * rsq * mask;
        q3[u] = (aS + aD * rsq) * mask;                 // d2Phi + (d-1)/r*dPhi
      }
      q4[u] = (aP + phib2) * mask;
    }
    v2f a;
    if (MODE == 0) {
      a.x = q1[0]; a.y = q1[1]; c1 = wmma_rowsum(a, ones, c1);
      a.x = q2[0]; a.y = q2[1]; c2 = wmma_rowsum(a, ones, c2);
      a.x = q3[0]; a.y = q3[1]; c3 = wmma_rowsum(a, ones, c3);
    }
    a.x = q4[0]; a.y = q4[1]; c4 = wmma_rowsum(a, ones, c4);
  }

  // C layout: vgpr g, lanes 0-15 -> row g; lanes 16-31 -> row g+8.
  // All columns identical, so lane 0 adds rows 0..7, lane 16 rows 8..15.
  if (irow == 0) {
    const int rbase = hv * 8;
#pragma unroll
    for (int g = 0; g < 8; ++g) {
      const int idx = b * Nn + it * 16 + rbase + g;
      if (MODE == 0) {
        atomic_addf(&ws[OFF_PS1 + idx], c1[g]);
        atomic_addf(&ws[OFF_PS2 + idx], c2[g]);
        atomic_addf(&ws[OFF_PS3 + idx], c3[g]);
        atomic_addf(&ws[OFF_PS4 + idx], c4[g]);
      } else {
        atomic_addf(&ws[OFF_PHIN + idx], c4[g]);
      }
    }
  }
}

// One thread per point: V (curr+next), grad V & lap V (curr only).
__global__ void onebody_kernel(const float* __restrict__ data,
                               const float* __restrict__ vw1,
                               const float* __restrict__ vb1,
                               const float* __restrict__ vw2,
                               const float* __restrict__ vb2,
                               float* __restrict__ ws) {
  const int tid = blockIdx.x * blockDim.x + threadIdx.x;
  if (tid >= 2 * BN) return;
  const int which = tid / BN;     // 0 = curr, 1 = next
  const int idx = tid % BN;
  const int b = idx / Nn, i = idx % Nn;
  const int m = b / (Ll - 1), lidx = b % (Ll - 1);
  const float* X = data + (size_t)(m * Ll + lidx + which) * Nn * Dd;
  const float x0 = X[i * 2 + 0], x1 = X[i * 2 + 1];

  float V = 0.0f, gx = 0.0f, gy = 0.0f, lap = 0.0f;
#pragma unroll 8
  for (int h = 0; h < Hh; ++h) {
    const float w0 = vw1[h];            // V_w1[0][h]
    const float w1 = vw1[Hh + h];       // V_w1[1][h]
    const float w2 = vw2[h];
    const float u = fmaf(x0, w0, fmaf(x1, w1, vb1[h]));
    const float t = TANHF(u);
    const float s = fmaf(-t, t, 1.0f);
    V = fmaf(w2, t, V);
    if (which == 0) {
      gx = fmaf(w2 * s, w0, gx);
      gy = fmaf(w2 * s, w1, gy);
      lap = fmaf(-2.0f * w2 * t * s, w0 * w0 + w1 * w1, lap);
    }
  }
  V += vb2[0];
  if (which == 0) {
    ws[OFF_VC + idx] = V;
    ws[OFF_GX + idx] = gx;
    ws[OFF_GY + idx] = gy;
    ws[OFF_LAPV + idx] = lap;
  } else {
    ws[OFF_VN + idx] = V;
  }
}

// Single block: assemble residual^2.
__global__ __launch_bounds__(256) void finish_kernel(
    const float* __restrict__ tsnap, const float* __restrict__ ws,
    float* __restrict__ out) {
  __shared__ float r1[256], r2[256], r3[256];
  const int tid = threadIdx.x;
  const float invN = 1.0f / (float)Nn;
  float s1 = 0.0f, s2 = 0.0f, s3 = 0.0f;
  for (int idx = tid; idx < BN; idx += 256) {
    const int b = idx / Nn;
    const int lidx = b % (Ll - 1);
    const float dt = tsnap[lidx + 1] - tsnap[lidx];
    // drift = -gradV - pairSum/N
    const float dxv = -ws[OFF_GX + idx] - ws[OFF_PS1 + idx] * invN;
    const float dyv = -ws[OFF_GY + idx] - ws[OFF_PS2 + idx] * invN;
    s1 += dt * invN * (dxv * dxv + dyv * dyv);                   // J_diss
    const float lap_i = ws[OFF_LAPV + idx] + ws[OFF_PS3 + idx] * invN;
    s2 += SIGMA2 * dt * invN * lap_i;                            // J_diff
    s3 += (ws[OFF_VN + idx] - ws[OFF_VC + idx]) * invN +
          (ws[OFF_PHIN + idx] - ws[OFF_PS4 + idx]) * invN * invN; // J_energy
  }
  r1[tid] = s1; r2[tid] = s2; r3[tid] = s3;
  __syncthreads();
  for (int off = 128; off > 0; off >>= 1) {
    if (tid < off) {
      r1[tid] += r1[tid + off];
      r2[tid] += r2[tid + off];
      r3[tid] += r3[tid + off];
    }
    __syncthreads();
  }
  if (tid == 0) {
    const float residual = (r1[0] + r2[0] - 2.0f * r3[0]) / (float)Bb;
    out[0] = residual * residual;
  }
}

extern "C" void kernel_launch(void* const* d_in, const int* in_sizes, int n_in,
                              void* d_out, int out_size, void* d_ws,
                              size_t ws_size, hipStream_t stream) {
  const float* data  = (const float*)d_in[0];
  const float* tsnap = (const float*)d_in[1];
  const float* vw1   = (const float*)d_in[2];
  const float* vb1   = (const float*)d_in[3];
  const float* vw2   = (const float*)d_in[4];
  const float* vb2   = (const float*)d_in[5];
  const float* pw1   = (const float*)d_in[6];
  const float* pb1   = (const float*)d_in[7];
  const float* pw2   = (const float*)d_in[8];
  const float* pb2   = (const float*)d_in[9];
  float* ws  = (float*)d_ws;
  float* out = (float*)d_out;

  // Zero the atomic accumulators (every call -> graph-replay deterministic).
  zero_kernel<<<dim3((ZERO_FLOATS + 255) / 256), dim3(256), 0, stream>>>(ws);
  // Pairwise passes: 20 b * 20 i-tiles * 4 j-segments single-wave workgroups.
  pair_kernel<0><<<dim3(Bb * NT, JS), dim3(32), 0, stream>>>(data, pw1, pb1,
                                                             pw2, pb2, ws);
  pair_kernel<1><<<dim3(Bb * NT, JS), dim3(32), 0, stream>>>(data, pw1, pb1,
                                                             pw2, pb2, ws);
  // One-body pass: 12800 points.
  onebody_kernel<<<dim3((2 * BN + 255) / 256), dim3(256), 0, stream>>>(
      data, vw1, vb1, vw2, vb2, ws);
  // Final assembly.
  finish_kernel<<<dim3(1), dim3(256), 0, stream>>>(tsnap, ws, out);
}